// H_derivatie_48069273977164
// MI455X (gfx1250) — compile-verified
//
#include <hip/hip_runtime.h>
#include <hip/hip_bf16.h>

typedef __attribute__((ext_vector_type(2))) float v2f;
typedef __attribute__((ext_vector_type(8))) float v8f;

#define NDIM 128

// ---------------------------------------------------------------- degree ----
__global__ void k_deg_init(float* deg, int n) {
    int i = blockIdx.x * blockDim.x + threadIdx.x;
    if (i < n) deg[i] = 1.0f;  // self-loop
}

__global__ void k_deg_count(const int* __restrict__ dst, float* deg, int e) {
    int i = blockIdx.x * blockDim.x + threadIdx.x;
    if (i < e) unsafeAtomicAdd(&deg[dst[i]], 1.0f);
}

__global__ void k_deg_rsqrt(float* deg, int n) {
    int i = blockIdx.x * blockDim.x + threadIdx.x;
    if (i < n) deg[i] = __frsqrt_rn(deg[i]);  // deg >= 1 always
}

// ------------------------------------------------------------- WMMA GEMM ----
// H[M x 128] = X[M x 128] @ W[128 x 128], fp32 via V_WMMA_F32_16X16X4_F32.
// Block = 256 threads (8 waves). Each block: 128 rows. Wave w: rows w*16..w*16+15,
// all 8 column tiles. W staged in LDS (64 KB).
__global__ void __launch_bounds__(256)
k_gemm128(const float* __restrict__ X, const float* __restrict__ W,
          float* __restrict__ H, int M) {
    __shared__ float Ws[NDIM * NDIM];  // 64 KB

    int tid = threadIdx.x;
    // cooperative stage of W into LDS (float4)
    {
        const float4* Wv = (const float4*)W;
        float4* Sv = (float4*)Ws;
        #pragma unroll
        for (int i = 0; i < 16; ++i)
            Sv[tid + 256 * i] = Wv[tid + 256 * i];  // 256*16*4 = 16384 floats
    }
    __syncthreads();

    int wave  = tid >> 5;
    int lane  = tid & 31;
    int l16   = lane & 15;
    int lhalf = lane >> 4;  // 0 or 1

    int rowBase = blockIdx.x * 128 + wave * 16;

    // A fragment row for this lane (clamped so OOB rows read valid memory;
    // their results are masked at store time)
    int arow = rowBase + l16;
    if (arow > M - 1) arow = M - 1;
    const float* aptr = X + (size_t)arow * NDIM + lhalf * 2;

    v8f acc[8] = {};

    for (int k = 0; k < NDIM; k += 4) {
        // A 16x4 fragment: VGPR0/1 = K = k + lhalf*2 + {0,1}
        v2f a;
        {
            const float2 av = *(const float2*)(aptr + k);
            a[0] = av.x; a[1] = av.y;
        }
        int krow = (k + lhalf * 2) * NDIM + l16;
        #pragma unroll
        for (int t = 0; t < 8; ++t) {
            // B 4x16 fragment for column tile t: lane -> N = t*16+l16,
            // VGPR0/1 = K = k + lhalf*2 + {0,1}
            v2f b;
            b[0] = Ws[krow + t * 16];
            b[1] = Ws[krow + NDIM + t * 16];
            acc[t] = __builtin_amdgcn_wmma_f32_16x16x4_f32(
                false, a, false, b, (short)0, acc[t], false, false);
        }
    }

    // C/D layout: VGPR v, lanes0-15 -> M=v, lanes16-31 -> M=v+8; N = lane&15
    #pragma unroll
    for (int t = 0; t < 8; ++t) {
        #pragma unroll
        for (int v = 0; v < 8; ++v) {
            int r = rowBase + v + lhalf * 8;
            if (r < M) H[(size_t)r * NDIM + t * 16 + l16] = acc[t][v];
        }
    }
}

// --------------------------------------------------- aggregation (scatter) ---
// out[i][:] = b[:] + h[i][:] * dinv[i]^2   (self-loop term + bias)
__global__ void k_agg_init(const float* __restrict__ h, const float* __restrict__ dinv,
                           const float* __restrict__ bias, float* __restrict__ out, int n) {
    int i = blockIdx.x * blockDim.x + threadIdx.x;  // over n*32, float4 each
    if (i >= n * 32) return;
    int node = i >> 5;
    int c4   = (i & 31) * 4;
    float di = dinv[node];
    float s  = di * di;
    float4 hv = *(const float4*)(h + (size_t)node * NDIM + c4);
    float4 bv = *(const float4*)(bias + c4);
    float4 o;
    o.x = bv.x + hv.x * s;
    o.y = bv.y + hv.y * s;
    o.z = bv.z + hv.z * s;
    o.w = bv.w + hv.w * s;
    *(float4*)(out + (size_t)node * NDIM + c4) = o;
}

// Edge scatter: 4 edges per wave, lane L handles channels 4L..4L+3 of each.
// Edge metadata fetched with wave-uniform (scalar) addresses; per-edge
// gathers issued up front so atomics of edge j overlap gathers of j+1.
// out[d][c] += h[s][c] * dinv[s]*dinv[d]  via hardware f32 atomics (L2-resident).
#define EPW 4
__global__ void __launch_bounds__(256)
k_agg_edges(const float* __restrict__ h, const float* __restrict__ dinv,
            const int* __restrict__ src, const int* __restrict__ dst,
            float* __restrict__ out, int e) {
    int wv   = __builtin_amdgcn_readfirstlane((int)(threadIdx.x >> 5));  // wave in block (SGPR)
    int lane = threadIdx.x & 31;
    int base = (blockIdx.x * 8 + wv) * EPW;
    if (base >= e) return;

    int    s[EPW], d[EPW];
    float  nrm[EPW];
    float4 hv[EPW];

    #pragma unroll
    for (int j = 0; j < EPW; ++j) {
        int eid = (base + j < e) ? (base + j) : (e - 1);  // clamp; dup is harmless? no -> guard below
        s[j] = src[eid];                                  // scalar loads (uniform addr)
        d[j] = dst[eid];
        nrm[j] = dinv[s[j]] * dinv[d[j]];
    }
    #pragma unroll
    for (int j = 0; j < EPW; ++j)
        hv[j] = *(const float4*)(h + (size_t)s[j] * NDIM + lane * 4);

    #pragma unroll
    for (int j = 0; j < EPW; ++j) {
        if (base + j >= e) break;   // only guard the side effects
        float* op = out + (size_t)d[j] * NDIM + lane * 4;
        unsafeAtomicAdd(op + 0, hv[j].x * nrm[j]);
        unsafeAtomicAdd(op + 1, hv[j].y * nrm[j]);
        unsafeAtomicAdd(op + 2, hv[j].z * nrm[j]);
        unsafeAtomicAdd(op + 3, hv[j].w * nrm[j]);
    }
}

__global__ void k_tanh_ip(float* p, int n4) {
    int i = blockIdx.x * blockDim.x + threadIdx.x;
    if (i >= n4) return;
    float4 v = ((float4*)p)[i];
    v.x = tanhf(v.x); v.y = tanhf(v.y); v.z = tanhf(v.z); v.w = tanhf(v.w);
    ((float4*)p)[i] = v;
}

// ------------------------------------------------------------------ launch ---
extern "C" void kernel_launch(void* const* d_in, const int* in_sizes, int n_in,
                              void* d_out, int out_size, void* d_ws, size_t ws_size,
                              hipStream_t stream) {
    const float* x   = (const float*)d_in[0];
    const float* W1  = (const float*)d_in[1];
    const float* b1  = (const float*)d_in[2];
    const float* W2  = (const float*)d_in[3];
    const float* b2  = (const float*)d_in[4];
    const int*   ei  = (const int*)d_in[5];

    const int N = in_sizes[0] / NDIM;   // 100000
    const int E = in_sizes[5] / 2;      // 600000
    const int* src = ei;
    const int* dst = ei + E;

    float* dinv = (float*)d_ws;                          // N floats (deg then dinv)
    float* h    = dinv + ((N + 255) & ~255);             // N*128 floats
    float* t1   = h + (size_t)N * NDIM;                  // N*128 floats
    float* outp = (float*)d_out;

    const int B = 256;
    int gN   = (N + B - 1) / B;
    int gE   = (E + B - 1) / B;
    int gN32 = (N * 32 + B - 1) / B;
    int gEw  = (E + 8 * EPW - 1) / (8 * EPW);   // 8 waves/block, EPW edges/wave
    int gM   = (N + 127) / 128;                 // GEMM row blocks

    // degrees (shared by both layers)
    k_deg_init <<<gN, B, 0, stream>>>(dinv, N);
    k_deg_count<<<gE, B, 0, stream>>>(dst, dinv, E);
    k_deg_rsqrt<<<gN, B, 0, stream>>>(dinv, N);

    // layer 1
    k_gemm128  <<<gM,   B, 0, stream>>>(x, W1, h, N);
    k_agg_init <<<gN32, B, 0, stream>>>(h, dinv, b1, t1, N);
    k_agg_edges<<<gEw,  B, 0, stream>>>(h, dinv, src, dst, t1, E);
    k_tanh_ip  <<<gN32, B, 0, stream>>>(t1, N * 32);

    // layer 2
    k_gemm128  <<<gM,   B, 0, stream>>>(t1, W2, h, N);
    k_agg_init <<<gN32, B, 0, stream>>>(h, dinv, b2, outp, N);
    k_agg_edges<<<gEw,  B, 0, stream>>>(h, dinv, src, dst, outp, E);
}